// LightGCNNetwork_79791902425516
// MI455X (gfx1250) — compile-verified
//
#include <hip/hip_runtime.h>
#include <stdint.h>

// ---------------------------------------------------------------------------
// LightGCN on MI455X (gfx1250, wave32).
// Memory-bound SpMM (edge table streamed via CDNA5 async-to-LDS, ASYNCcnt),
// 38.4MB embedding table is L2-resident (192MB L2), hardware f32 atomics
// scatter (no-return -> STOREcnt). WMMA is inapplicable: per-row neighbor
// sets share no K operand, so the workload has no dense matmul structure.
// ---------------------------------------------------------------------------

#define DIM    64
#define DV     16          // float4 quads per embedding row
#define EPB    1024        // edges per block tile (12KB LDS stage)
#define TPB    256         // 8 waves of 32

// Guarantee the hardware global_atomic_add_f32 (not a CAS loop).
__device__ __forceinline__ void atomic_add_f32(float* p, float v) {
#if defined(__gfx1250__)
    unsafeAtomicAdd(p, v);
#else
    atomicAdd(p, v);
#endif
}

// ---- SpMM: embout[rows[e]] += vals[e] * embin[cols[e]] -------------------
__global__ __launch_bounds__(TPB)
void lgcn_spmm(const int* __restrict__ rows,
               const int* __restrict__ cols,
               const float* __restrict__ vals,
               const float* __restrict__ embin,
               float* __restrict__ embout,
               int nnz)
{
    __shared__ int   sh_rows[EPB];
    __shared__ int   sh_cols[EPB];
    __shared__ float sh_vals[EPB];

    const int  tid  = threadIdx.x;                 // 0..255
    const long base = (long)blockIdx.x * EPB;

    // ---- Stage this tile's edge data into LDS with CDNA5 async loads.
    // Each lane moves 16B of each array (256 lanes x 4 elems = 1024 elems).
    // Lanes whose 16B chunk starts in-bounds issue; a <=12B overread at the
    // very end of the arrays stays within allocation granularity.
    {
        const long g = base + (long)tid * 4;
        if (g < (long)nnz) {
            uint64_t ga;
            uint32_t la;   // generic LDS pointer low 32 bits == LDS byte addr
            ga = (uint64_t)(uintptr_t)(rows + g);
            la = (uint32_t)(uintptr_t)&sh_rows[tid * 4];
            asm volatile("global_load_async_to_lds_b128 %0, %1, off"
                         :: "v"(la), "v"(ga) : "memory");
            ga = (uint64_t)(uintptr_t)(cols + g);
            la = (uint32_t)(uintptr_t)&sh_cols[tid * 4];
            asm volatile("global_load_async_to_lds_b128 %0, %1, off"
                         :: "v"(la), "v"(ga) : "memory");
            ga = (uint64_t)(uintptr_t)(vals + g);
            la = (uint32_t)(uintptr_t)&sh_vals[tid * 4];
            asm volatile("global_load_async_to_lds_b128 %0, %1, off"
                         :: "v"(la), "v"(ga) : "memory");
        }
        asm volatile("s_wait_asynccnt 0" ::: "memory");
    }
    __syncthreads();

    const long rem    = (long)nnz - base;
    const int  tile_n = rem < (long)EPB ? (int)rem : EPB;

    // 16 lanes cooperate on one edge: lane quad q covers dims [4q,4q+4).
    const int lane16 = tid & 15;
    const int sub    = tid >> 4;                   // 16 edge slots per pass
    const float4* embin4 = (const float4*)embin;

    for (int e = sub; e < tile_n; e += 16) {
        const int   r = sh_rows[e];
        const int   c = sh_cols[e];
        const float v = sh_vals[e];

        // Prefetch next edge's gather row into cache (global_prefetch_b8).
        const int en = e + 16;
        if (en < tile_n)
            __builtin_prefetch(&embin[(long)sh_cols[en] * DIM], 0, 0);

        const float4 g = embin4[(long)c * DV + lane16];   // global_load_b128
        float* dst = embout + (long)r * DIM + lane16 * 4;
        atomic_add_f32(dst + 0, v * g.x);                 // global_atomic_add_f32
        atomic_add_f32(dst + 1, v * g.y);
        atomic_add_f32(dst + 2, v * g.z);
        atomic_add_f32(dst + 3, v * g.w);
    }
}

// ---- acc = w ; zbuf = 0 ---------------------------------------------------
__global__ void lgcn_init(const float* __restrict__ w,
                          float* __restrict__ acc,
                          float* __restrict__ zbuf, int n4)
{
    int i = blockIdx.x * blockDim.x + threadIdx.x;
    if (i < n4) {
        ((float4*)acc)[i]  = ((const float4*)w)[i];
        ((float4*)zbuf)[i] = make_float4(0.f, 0.f, 0.f, 0.f);
    }
}

// ---- acc += src ; optionally zbuf = 0 ------------------------------------
__global__ void lgcn_accum(float* __restrict__ acc,
                           const float* __restrict__ src,
                           float* __restrict__ zbuf, int n4)
{
    int i = blockIdx.x * blockDim.x + threadIdx.x;
    if (i < n4) {
        float4 a = ((float4*)acc)[i];
        float4 s = ((const float4*)src)[i];
        a.x += s.x; a.y += s.y; a.z += s.z; a.w += s.w;
        ((float4*)acc)[i] = a;
        if (zbuf) ((float4*)zbuf)[i] = make_float4(0.f, 0.f, 0.f, 0.f);
    }
}

// ---- final gathers: six (B,64) sections ----------------------------------
__global__ void lgcn_gather(const float* __restrict__ w,
                            const float* __restrict__ acc,
                            const int* __restrict__ uidx,
                            const int* __restrict__ pidx,
                            const int* __restrict__ nidx,
                            const int* __restrict__ num_users_p,
                            const int* __restrict__ num_layers_p,
                            float* __restrict__ out, int Bn)
{
    int t = blockIdx.x * blockDim.x + threadIdx.x;      // over Bn * DV
    if (t >= Bn * DV) return;
    const int b = t / DV;
    const int q = t % DV;
    const int U = num_users_p[0];
    const float inv = 1.0f / (float)(num_layers_p[0] + 1);

    const float4* w4 = (const float4*)w;
    const float4* a4 = (const float4*)acc;
    float4*       o4 = (float4*)out;
    const long sec = (long)Bn * DV;
    const long bq  = (long)b * DV + q;

    const int ui = uidx[b], pi = pidx[b], ni = nidx[b];

    float4 v;
    v = a4[(long)ui * DV + q];
    v.x *= inv; v.y *= inv; v.z *= inv; v.w *= inv;
    o4[0 * sec + bq] = v;

    v = a4[((long)U + pi) * DV + q];
    v.x *= inv; v.y *= inv; v.z *= inv; v.w *= inv;
    o4[1 * sec + bq] = v;

    v = a4[((long)U + ni) * DV + q];
    v.x *= inv; v.y *= inv; v.z *= inv; v.w *= inv;
    o4[2 * sec + bq] = v;

    // Reference quirk: all three 0-emb outputs index embed_weight's USER rows,
    // including with item indices (init_user[pos_item_idxs]). Reproduce it.
    o4[3 * sec + bq] = w4[(long)ui * DV + q];
    o4[4 * sec + bq] = w4[(long)pi * DV + q];
    o4[5 * sec + bq] = w4[(long)ni * DV + q];
}

extern "C" void kernel_launch(void* const* d_in, const int* in_sizes, int n_in,
                              void* d_out, int out_size, void* d_ws, size_t ws_size,
                              hipStream_t stream)
{
    const float* w     = (const float*)d_in[0];   // (N, 64)
    const float* vals  = (const float*)d_in[1];   // (nnz,)
    const int*   rows  = (const int*)  d_in[2];   // (nnz,)
    const int*   cols  = (const int*)  d_in[3];   // (nnz,)
    const int*   uidx  = (const int*)  d_in[4];   // (B,)
    const int*   pidx  = (const int*)  d_in[5];
    const int*   nidx  = (const int*)  d_in[6];
    const int*   nup   = (const int*)  d_in[7];   // num_users (device scalar)
    const int*   nlay  = (const int*)  d_in[9];   // num_layers (device scalar)

    const int nnz = in_sizes[2];
    const int NE  = in_sizes[0];                  // N * 64 elements
    const int n4  = NE / 4;                       // float4 count
    const int Bn  = in_sizes[4];

    float* emb_a = (float*)d_ws;                  // N*64 floats
    float* emb_b = emb_a + NE;                    // N*64 floats
    float* acc   = emb_b + NE;                    // N*64 floats

    const int eblk = (nnz + EPB - 1) / EPB;
    const int vblk = (n4 + TPB - 1) / TPB;
    const int gblk = (Bn * DV + TPB - 1) / TPB;

    // acc = W ; emb_b = 0
    lgcn_init<<<vblk, TPB, 0, stream>>>(w, acc, emb_b, n4);

    // Layer 1: emb_b = A * W ; acc += emb_b ; emb_a = 0
    lgcn_spmm<<<eblk, TPB, 0, stream>>>(rows, cols, vals, w, emb_b, nnz);
    lgcn_accum<<<vblk, TPB, 0, stream>>>(acc, emb_b, emb_a, n4);

    // Layer 2: emb_a = A * emb_b ; acc += emb_a ; emb_b = 0
    lgcn_spmm<<<eblk, TPB, 0, stream>>>(rows, cols, vals, emb_b, emb_a, nnz);
    lgcn_accum<<<vblk, TPB, 0, stream>>>(acc, emb_a, emb_b, n4);

    // Layer 3: emb_b = A * emb_a ; acc += emb_b
    lgcn_spmm<<<eblk, TPB, 0, stream>>>(rows, cols, vals, emb_a, emb_b, nnz);
    lgcn_accum<<<vblk, TPB, 0, stream>>>(acc, emb_b, nullptr, n4);

    // Mean + six gathers
    lgcn_gather<<<gblk, TPB, 0, stream>>>(w, acc, uidx, pidx, nidx, nup, nlay,
                                          (float*)d_out, Bn);
}